// ICNNPolicy_7421703487770
// MI455X (gfx1250) — compile-verified
//
#include <hip/hip_runtime.h>
#include <math.h>

// ICNN policy on MI455X (gfx1250): all GEMMs via V_WMMA_F32_16X16X4_F32 (fp32 exact).
// One workgroup = 8 wave32 = 32-sample batch tile (2 WMMA N-tiles per A-fragment,
// halving L2 traffic on the dominant 512x512 softplus(Wz1) GEMMs). H=512 -> 32 M-tiles.

#define BTOT   16384
#define NX     12
#define NU     5
#define HDIM   512
#define DTC    0.02f
#define GAMMAC 0.99f
#define TN     32            // samples per workgroup (two 16-wide N tiles)
#define HHALF  (512*16)      // half-buffer stride for 512-row activations

typedef __attribute__((ext_vector_type(2))) float v2f;
typedef __attribute__((ext_vector_type(8))) float v8f;

__device__ __forceinline__ v8f wmma4(v2f a, v2f b, v8f c) {
  // D = A(16x4 f32) * B(4x16 f32) + C(16x16 f32)
  return __builtin_amdgcn_wmma_f32_16x16x4_f32(
      /*neg_a=*/false, a, /*neg_b=*/false, b,
      /*c_mod=*/(short)0, c, /*reuse_a=*/false, /*reuse_b=*/false);
}

__device__ __forceinline__ float softplusf(float xv) {
  return (xv > 20.f) ? xv : log1pf(expf(xv));
}

// Activation layout: two independent 16-column half-buffers (row stride 16 each)
// -> conflict-free LDS B-fragment loads. n in [0,32): half = n>>4, col = n&15.
__device__ __forceinline__ int aidx(int rows, int m, int n) {
  return ((n >> 4) * rows + m) * 16 + (n & 15);
}

// Precompute softplus(Wz1) (row-major + transposed) and softplus(wz2).
__global__ void icnn_prep_kernel(const float* __restrict__ Wz1,
                                 const float* __restrict__ wz2,
                                 float* __restrict__ sp1,
                                 float* __restrict__ sp1T,
                                 float* __restrict__ spwz2) {
  int stride = gridDim.x * blockDim.x;
  int total = HDIM * HDIM + HDIM;
  for (int i = blockIdx.x * blockDim.x + threadIdx.x; i < total; i += stride) {
    if (i < HDIM * HDIM) {
      float s = softplusf(Wz1[i]);
      sp1[i] = s;
      sp1T[(i & (HDIM - 1)) * HDIM + (i >> 9)] = s;   // [c][r] = [r][c]
    } else {
      spwz2[i - HDIM * HDIM] = softplusf(wz2[i - HDIM * HDIM]);
    }
  }
}

__global__ __launch_bounds__(256) void icnn_policy_kernel(
    const float* __restrict__ x,     const float* __restrict__ x_ref,
    const float* __restrict__ u_nom, const float* __restrict__ tilt,
    const float* __restrict__ W1,    const float* __restrict__ b1,
    const float* __restrict__ W2,    const float* __restrict__ b2,
    const float* __restrict__ Wy0,   const float* __restrict__ by0,
    const float* __restrict__ sp1,   const float* __restrict__ sp1T,
    const float* __restrict__ Wy1,   const float* __restrict__ by1,
    const float* __restrict__ spwz2, const float* __restrict__ wy2,
    const float* __restrict__ r_diag, const float* __restrict__ u_min,
    const float* __restrict__ u_max, float* __restrict__ out)
{
  extern __shared__ float smem[];
  float* sInp  = smem;              // 2 x (20x16)
  float* sY    = sInp  + 2*20*16;   // 2 x (24x16) : y = [x_next; x_ref]
  float* sG    = sY    + 2*24*16;   // 2 x (16x16) : grad_V (rows 0..11 valid)
  float* sPart = sG    + 2*16*16;   // 8 x (16x32) : cross-wave K-split partials
  float* sH    = sPart + 8*512;     // 2 x (512x16): h = tanh(.)
  float* sZ1   = sH    + 2*HHALF;   // 2 x (512x16): z1 = softplus(.)
  float* sT2   = sZ1   + 2*HHALF;   // 2 x (512x16): t2 (later reused as r)
  float* sT1   = sT2   + 2*HHALF;   // 2 x (512x16): t1

  const int tid  = threadIdx.x;
  const int wave = tid >> 5;        // 0..7
  const int lane = tid & 31;
  const int lh   = lane >> 4;       // K-pair / M+8 half select
  const int lm   = lane & 15;       // row/col within 16
  const int base_n = blockIdx.x * TN;

  // ---- Stage 0: build inp = [x; u_nom; tilt; 0] (20 x 32) ----
  for (int t = tid; t < 20 * TN; t += 256) {
    int k = t >> 5, n = t & 31;
    int s = base_n + n;
    float v;
    if (k < 12)      v = x[s * NX + k];
    else if (k < 17) v = u_nom[s * NU + (k - 12)];
    else if (k < 19) v = tilt[s * 2 + (k - 17)];
    else             v = 0.f;
    sInp[aidx(20, k, n)] = v;
  }
  __syncthreads();

  // ---- Stage 1: h = tanh(W1 @ inp + b1)   (512x19)*(19x32) ----
  for (int mt = 0; mt < 4; ++mt) {
    int mbase = (wave * 4 + mt) * 16;
    int row = mbase + lm;
    v8f acc0 = {}, acc1 = {};
#pragma unroll
    for (int ks = 0; ks < 5; ++ks) {
      int k = ks * 4 + 2 * lh;
      float mx = (k     < 19) ? 1.f : 0.f;  int kx = (k     < 19) ? k     : 0;
      float my = (k + 1 < 19) ? 1.f : 0.f;  int ky = (k + 1 < 19) ? k + 1 : 0;
      v2f a, b0, b1f;
      a.x = W1[row * 19 + kx] * mx;
      a.y = W1[row * 19 + ky] * my;
      b0.x  = sInp[k * 16 + lm];          b0.y  = sInp[(k + 1) * 16 + lm];
      b1f.x = sInp[320 + k * 16 + lm];    b1f.y = sInp[320 + (k + 1) * 16 + lm];
      acc0 = wmma4(a, b0, acc0);
      acc1 = wmma4(a, b1f, acc1);
    }
#pragma unroll
    for (int v = 0; v < 8; ++v) {
      int m = mbase + v + 8 * lh;
      sH[m * 16 + lm]         = tanhf(acc0[v] + b1[m]);
      sH[HHALF + m * 16 + lm] = tanhf(acc1[v] + b1[m]);
    }
  }
  __syncthreads();

  // ---- Stage 2: x_next = x + DT*(W2 @ h + b2); waves split K=512 ----
  {
    float mk = (lm < 12) ? 1.f : 0.f;
    int rowc = (lm < 12) ? lm : 0;
    v8f acc0 = {}, acc1 = {};
    for (int ks = 0; ks < 16; ++ks) {
      int k = wave * 64 + ks * 4 + 2 * lh;
      v2f a, b0, b1f;
      a.x = W2[rowc * HDIM + k] * mk;
      a.y = W2[rowc * HDIM + k + 1] * mk;
      b0.x  = sH[k * 16 + lm];          b0.y  = sH[(k + 1) * 16 + lm];
      b1f.x = sH[HHALF + k * 16 + lm];  b1f.y = sH[HHALF + (k + 1) * 16 + lm];
      acc0 = wmma4(a, b0, acc0);
      acc1 = wmma4(a, b1f, acc1);
    }
#pragma unroll
    for (int v = 0; v < 8; ++v) {
      sPart[wave * 512 + (v + 8 * lh) * 16 + lm]       = acc0[v];
      sPart[wave * 512 + 256 + (v + 8 * lh) * 16 + lm] = acc1[v];
    }
  }
  __syncthreads();
  for (int t = tid; t < 16 * TN; t += 256) {
    int m = t >> 5, n = t & 31;
    if (m < 12) {
      float s = 0.f;
      for (int w = 0; w < 8; ++w) s += sPart[w * 512 + aidx(16, m, n)];
      int smp = base_n + n;
      sY[aidx(24, m, n)]      = x[smp * NX + m] + DTC * (s + b2[m]);
      sY[aidx(24, 12 + m, n)] = x_ref[smp * NX + m];
    }
  }
  __syncthreads();

  // ---- Stage 3: z1 = softplus(Wy0 @ y + by0)   (512x24)*(24x32) ----
  for (int mt = 0; mt < 4; ++mt) {
    int mbase = (wave * 4 + mt) * 16;
    int row = mbase + lm;
    v8f acc0 = {}, acc1 = {};
#pragma unroll
    for (int ks = 0; ks < 6; ++ks) {
      int k = ks * 4 + 2 * lh;
      v2f a, b0, b1f;
      a.x = Wy0[row * 24 + k];
      a.y = Wy0[row * 24 + k + 1];
      b0.x  = sY[k * 16 + lm];          b0.y  = sY[(k + 1) * 16 + lm];
      b1f.x = sY[384 + k * 16 + lm];    b1f.y = sY[384 + (k + 1) * 16 + lm];
      acc0 = wmma4(a, b0, acc0);
      acc1 = wmma4(a, b1f, acc1);
    }
#pragma unroll
    for (int v = 0; v < 8; ++v) {
      int m = mbase + v + 8 * lh;
      sZ1[m * 16 + lm]         = softplusf(acc0[v] + by0[m]);
      sZ1[HHALF + m * 16 + lm] = softplusf(acc1[v] + by0[m]);
    }
  }
  __syncthreads();

  // ---- Stage 4: t2 = softplus(wz2) * sigmoid(sp(Wz1)@z1 + Wy1@y + by1) ----
  for (int mt = 0; mt < 4; ++mt) {
    int mbase = (wave * 4 + mt) * 16;
    int row = mbase + lm;
    v8f acc0 = {}, acc1 = {};
#pragma unroll
    for (int ks = 0; ks < 6; ++ks) {
      int k = ks * 4 + 2 * lh;
      v2f a, b0, b1f;
      a.x = Wy1[row * 24 + k];
      a.y = Wy1[row * 24 + k + 1];
      b0.x  = sY[k * 16 + lm];          b0.y  = sY[(k + 1) * 16 + lm];
      b1f.x = sY[384 + k * 16 + lm];    b1f.y = sY[384 + (k + 1) * 16 + lm];
      acc0 = wmma4(a, b0, acc0);
      acc1 = wmma4(a, b1f, acc1);
    }
    for (int ks = 0; ks < 128; ++ks) {   // big forward GEMM, K=512
      int k = ks * 4 + 2 * lh;
      v2f a, b0, b1f;
      a.x = sp1[row * HDIM + k];
      a.y = sp1[row * HDIM + k + 1];
      b0.x  = sZ1[k * 16 + lm];          b0.y  = sZ1[(k + 1) * 16 + lm];
      b1f.x = sZ1[HHALF + k * 16 + lm];  b1f.y = sZ1[HHALF + (k + 1) * 16 + lm];
      acc0 = wmma4(a, b0, acc0);
      acc1 = wmma4(a, b1f, acc1);
    }
#pragma unroll
    for (int v = 0; v < 8; ++v) {
      int m = mbase + v + 8 * lh;
      float spz = spwz2[m];
      float z2p0 = acc0[v] + by1[m];
      float z2p1 = acc1[v] + by1[m];
      sT2[m * 16 + lm]         = spz / (1.f + expf(-z2p0));
      sT2[HHALF + m * 16 + lm] = spz / (1.f + expf(-z2p1));
    }
  }
  __syncthreads();

  // ---- Stage 5: t1 = sigmoid(z1pre) * (sp(Wz1)^T @ t2); sigmoid = 1-exp(-z1) ----
  for (int mt = 0; mt < 4; ++mt) {
    int mbase = (wave * 4 + mt) * 16;
    int row = mbase + lm;
    v8f acc0 = {}, acc1 = {};
    for (int ks = 0; ks < 128; ++ks) {   // big backward GEMM, K=512
      int k = ks * 4 + 2 * lh;
      v2f a, b0, b1f;
      a.x = sp1T[row * HDIM + k];        // row-contiguous: merges to b64
      a.y = sp1T[row * HDIM + k + 1];
      b0.x  = sT2[k * 16 + lm];          b0.y  = sT2[(k + 1) * 16 + lm];
      b1f.x = sT2[HHALF + k * 16 + lm];  b1f.y = sT2[HHALF + (k + 1) * 16 + lm];
      acc0 = wmma4(a, b0, acc0);
      acc1 = wmma4(a, b1f, acc1);
    }
#pragma unroll
    for (int v = 0; v < 8; ++v) {
      int m = mbase + v + 8 * lh;
      float sg0 = 1.f - expf(-sZ1[m * 16 + lm]);
      float sg1 = 1.f - expf(-sZ1[HHALF + m * 16 + lm]);
      sT1[m * 16 + lm]         = sg0 * acc0[v];
      sT1[HHALF + m * 16 + lm] = sg1 * acc1[v];
    }
  }
  __syncthreads();

  // ---- Stage 6: grad_V[0:12] = Wy0[:,:12]^T@t1 + Wy1[:,:12]^T@t2 + wy2[:12] ----
  {
    v8f acc0 = {}, acc1 = {};
    for (int ks = 0; ks < 16; ++ks) {
      int k = wave * 64 + ks * 4 + 2 * lh;
      v2f a, b0, b1f;
      a.x = Wy0[k * 24 + lm];            // lm 12..15 -> unused garbage rows
      a.y = Wy0[(k + 1) * 24 + lm];
      b0.x  = sT1[k * 16 + lm];          b0.y  = sT1[(k + 1) * 16 + lm];
      b1f.x = sT1[HHALF + k * 16 + lm];  b1f.y = sT1[HHALF + (k + 1) * 16 + lm];
      acc0 = wmma4(a, b0, acc0);
      acc1 = wmma4(a, b1f, acc1);
      a.x = Wy1[k * 24 + lm];
      a.y = Wy1[(k + 1) * 24 + lm];
      b0.x  = sT2[k * 16 + lm];          b0.y  = sT2[(k + 1) * 16 + lm];
      b1f.x = sT2[HHALF + k * 16 + lm];  b1f.y = sT2[HHALF + (k + 1) * 16 + lm];
      acc0 = wmma4(a, b0, acc0);
      acc1 = wmma4(a, b1f, acc1);
    }
#pragma unroll
    for (int v = 0; v < 8; ++v) {
      sPart[wave * 512 + (v + 8 * lh) * 16 + lm]       = acc0[v];
      sPart[wave * 512 + 256 + (v + 8 * lh) * 16 + lm] = acc1[v];
    }
  }
  __syncthreads();
  for (int t = tid; t < 16 * TN; t += 256) {
    int m = t >> 5, n = t & 31;
    float s = 0.f;
    for (int w = 0; w < 8; ++w) s += sPart[w * 512 + aidx(16, m, n)];
    sG[aidx(16, m, n)] = (m < 12) ? (s + wy2[m]) : 0.f;
  }
  __syncthreads();

  // ---- Stage 7: r = (1-h^2) * (W2^T @ grad_V)   (512x12)*(12x32); r -> sT2 ----
  for (int mt = 0; mt < 4; ++mt) {
    int mbase = (wave * 4 + mt) * 16;
    int col = mbase + lm;
    v8f acc0 = {}, acc1 = {};
#pragma unroll
    for (int ks = 0; ks < 3; ++ks) {
      int k = ks * 4 + 2 * lh;           // K = 12 exactly
      v2f a, b0, b1f;
      a.x = W2[k * HDIM + col];
      a.y = W2[(k + 1) * HDIM + col];
      b0.x  = sG[k * 16 + lm];           b0.y  = sG[(k + 1) * 16 + lm];
      b1f.x = sG[256 + k * 16 + lm];     b1f.y = sG[256 + (k + 1) * 16 + lm];
      acc0 = wmma4(a, b0, acc0);
      acc1 = wmma4(a, b1f, acc1);
    }
#pragma unroll
    for (int v = 0; v < 8; ++v) {
      int m = mbase + v + 8 * lh;
      float h0 = sH[m * 16 + lm];
      float h1 = sH[HHALF + m * 16 + lm];
      sT2[m * 16 + lm]         = acc0[v] * (1.f - h0 * h0);
      sT2[HHALF + m * 16 + lm] = acc1[v] * (1.f - h1 * h1);
    }
  }
  __syncthreads();

  // ---- Stage 8: g = GAMMA*DT * W1[:,12:17]^T @ r; u = clip(-g/r_diag) ----
  {
    float mk = (lm < NU) ? 1.f : 0.f;
    int cc = (lm < NU) ? lm : 0;
    v8f acc0 = {}, acc1 = {};
    for (int ks = 0; ks < 16; ++ks) {
      int k = wave * 64 + ks * 4 + 2 * lh;
      v2f a, b0, b1f;
      a.x = W1[k * 19 + 12 + cc] * mk;
      a.y = W1[(k + 1) * 19 + 12 + cc] * mk;
      b0.x  = sT2[k * 16 + lm];          b0.y  = sT2[(k + 1) * 16 + lm];
      b1f.x = sT2[HHALF + k * 16 + lm];  b1f.y = sT2[HHALF + (k + 1) * 16 + lm];
      acc0 = wmma4(a, b0, acc0);
      acc1 = wmma4(a, b1f, acc1);
    }
#pragma unroll
    for (int v = 0; v < 8; ++v) {
      sPart[wave * 512 + (v + 8 * lh) * 16 + lm]       = acc0[v];
      sPart[wave * 512 + 256 + (v + 8 * lh) * 16 + lm] = acc1[v];
    }
  }
  __syncthreads();
  if (tid < NU * TN) {
    int m = tid >> 5, n = tid & 31;     // m in 0..4, n in 0..31
    float s = 0.f;
    for (int w = 0; w < 8; ++w) s += sPart[w * 512 + aidx(16, m, n)];
    float g = GAMMAC * DTC * s;
    float u = -g / r_diag[m];
    u = fminf(fmaxf(u, u_min[m]), u_max[m]);
    out[(base_n + n) * NU + m] = u;
  }
}

extern "C" void kernel_launch(void* const* d_in, const int* in_sizes, int n_in,
                              void* d_out, int out_size, void* d_ws, size_t ws_size,
                              hipStream_t stream) {
  const float* x      = (const float*)d_in[0];
  const float* x_ref  = (const float*)d_in[1];
  const float* u_nom  = (const float*)d_in[2];
  const float* tilt   = (const float*)d_in[3];
  const float* W1     = (const float*)d_in[4];
  const float* b1     = (const float*)d_in[5];
  const float* W2     = (const float*)d_in[6];
  const float* b2     = (const float*)d_in[7];
  const float* Wy0    = (const float*)d_in[8];
  const float* by0    = (const float*)d_in[9];
  const float* Wz1    = (const float*)d_in[10];
  const float* Wy1    = (const float*)d_in[11];
  const float* by1    = (const float*)d_in[12];
  const float* wz2    = (const float*)d_in[13];
  const float* wy2    = (const float*)d_in[14];
  // d_in[15] = bV : constant offset, vanishes in grad_V -> unused
  const float* r_diag = (const float*)d_in[16];
  const float* u_min  = (const float*)d_in[17];
  const float* u_max  = (const float*)d_in[18];

  float* sp1   = (float*)d_ws;            // 512*512 floats
  float* sp1T  = sp1 + HDIM * HDIM;       // 512*512 floats (transposed copy)
  float* spwz2 = sp1T + HDIM * HDIM;      // 512 floats

  icnn_prep_kernel<<<512, 256, 0, stream>>>(Wz1, wz2, sp1, sp1T, spwz2);

  size_t lds_bytes = (size_t)(2*20*16 + 2*24*16 + 2*16*16 + 8*512 + 4*2*HHALF)
                     * sizeof(float);     // ~286 KB, fits 320 KB/WGP
  icnn_policy_kernel<<<BTOT / TN, 256, lds_bytes, stream>>>(
      x, x_ref, u_nom, tilt, W1, b1, W2, b2, Wy0, by0, sp1, sp1T, Wy1, by1,
      spwz2, wy2, r_diag, u_min, u_max, (float*)d_out);
}